// BuddingLayer_53661321396545
// MI455X (gfx1250) — compile-verified
//
#include <hip/hip_runtime.h>

// BuddingLayer for MI455X (gfx1250, wave32).
// Memory-bound problem (~768MB effective traffic @23.3TB/s ~= 33us floor):
//  - prep: collapse per-neuron 3x3 MLPs to h2[k] (3 floats) + mask, and x_masked.
//  - buds: u[o] = sum_k mask[k]*relu(W3[k,o,:].h2[k] + b3[k,o]); mask=0 rows are
//    skipped (block-uniform branch) -> ~halves the 1GB W3/b3 stream.
//  - gemv: weight @ x_masked via V_WMMA_F32_16X16X4_F32, C-register chained over k.
//  - reduce: deterministic fixed-order sum of split-k partials + bias.

#define N   8192
#define KC  32    // bud split-k chunks (N/KC = 256 k per block)
#define KS  16    // gemv split-k chunks (N/KS = 512 k per wave)

typedef __attribute__((ext_vector_type(2))) float v2f;
typedef __attribute__((ext_vector_type(8))) float v8f;

struct F3 { float x, y, z; };

// ---------------------------------------------------------------- prep
__global__ void __launch_bounds__(256)
prep_kernel(const float* __restrict__ x, const int* __restrict__ sat,
            const float* __restrict__ W1, const float* __restrict__ b1,
            const float* __restrict__ W2, const float* __restrict__ b2,
            float4* __restrict__ h2m, float* __restrict__ xm) {
  int k = blockIdx.x * blockDim.x + threadIdx.x;
  if (k >= N) return;
  float xv = x[k];
  int  s  = sat[k];
  float satf = s ? 1.0f : 0.0f;
  xm[k] = xv * (1.0f - satf);
  float maskf = (s && xv != 0.0f) ? 1.0f : 0.0f;

  // h0 = x/3 replicated 3x  =>  h1[o] = relu((x/3) * rowsum(W1[o,:]) + b1[o])
  float h0 = xv * (1.0f / 3.0f);
  const float* w1 = W1 + (size_t)k * 9;
  const float* B1 = b1 + (size_t)k * 3;
  float h1_0 = fmaxf(fmaf(h0, w1[0] + w1[1] + w1[2], B1[0]), 0.0f);
  float h1_1 = fmaxf(fmaf(h0, w1[3] + w1[4] + w1[5], B1[1]), 0.0f);
  float h1_2 = fmaxf(fmaf(h0, w1[6] + w1[7] + w1[8], B1[2]), 0.0f);

  const float* w2 = W2 + (size_t)k * 9;
  const float* B2 = b2 + (size_t)k * 3;
  float h2_0 = fmaxf(fmaf(w2[0], h1_0, fmaf(w2[1], h1_1, fmaf(w2[2], h1_2, B2[0]))), 0.0f);
  float h2_1 = fmaxf(fmaf(w2[3], h1_0, fmaf(w2[4], h1_1, fmaf(w2[5], h1_2, B2[1]))), 0.0f);
  float h2_2 = fmaxf(fmaf(w2[6], h1_0, fmaf(w2[7], h1_1, fmaf(w2[8], h1_2, B2[2]))), 0.0f);

  h2m[k] = make_float4(h2_0, h2_1, h2_2, maskf);
}

// ---------------------------------------------------------------- buds
// grid = (N/256 o-tiles, KC k-chunks), block = 256 threads (one o each).
// W3 access: consecutive threads -> 12B stride, fully covered cache lines.
// Base pointers are bumped per-k (constant delta) instead of remultiplying
// (size_t)k*N each iteration.
__global__ void __launch_bounds__(256)
buds_kernel(const float* __restrict__ W3, const float* __restrict__ b3,
            const float4* __restrict__ h2m, float* __restrict__ budp) {
  int o    = blockIdx.x * blockDim.x + threadIdx.x;
  int kbeg = blockIdx.y * (N / KC);
  const float*  w3p = W3 + ((size_t)kbeg * N + o) * 3;
  const float*  b3p = b3 + ((size_t)kbeg * N + o);
  const float4* hp  = h2m + kbeg;
  float acc = 0.0f;
  for (int i = 0; i < N / KC; ++i) {
    float4 h = hp[i];                   // uniform address: scalar-cached
    if (h.w != 0.0f) {                  // block-uniform row skip (~50% traffic cut)
      const F3 w  = *(const F3*)w3p;
      float  pre  = fmaf(w.x, h.x, fmaf(w.y, h.y, fmaf(w.z, h.z, *b3p)));
      acc += fmaxf(pre, 0.0f);
    }
    w3p += (size_t)3 * N;
    b3p += N;
  }
  budp[(size_t)blockIdx.y * N + o] = acc;
}

// ---------------------------------------------------------------- gemv (WMMA)
// One wave32 per block so EXEC is all-1s at the WMMA.
// A (16x4 f32, 2 VGPRs): lane<16 holds {K0,K1} of row m=lane,
//                        lane>=16 holds {K2,K3} of row m=lane-16.
// B (4x16): columns broadcast -> VGPR0 = x[k+2*sel], VGPR1 = x[k+2*sel+1].
// C chained across the k loop does the split-k accumulation in-register.
// Bases hoisted so the kk deltas (<= 2KB) fold into instruction IOFFSETs.
__global__ void __launch_bounds__(32)
gemv_kernel(const float* __restrict__ weight, const float* __restrict__ xm,
            float* __restrict__ gvp) {
  int lane = threadIdx.x;
  int sel  = lane >> 4;          // 0: K0/K1 half, 1: K2/K3 half
  int m    = lane & 15;
  int o0   = blockIdx.x * 16;
  int kbase = blockIdx.y * (N / KS);

  const float* wp = weight + (size_t)(o0 + m) * N + kbase + 2 * sel;
  const float* xp = xm + kbase + 2 * sel;

  v8f c = {};
  #pragma unroll 8
  for (int kk = 0; kk < N / KS; kk += 4) {
    v2f a = *(const v2f*)(wp + kk);    // 8B-aligned (base even, kk multiple of 4)
    v2f b;
    b.x = xp[kk];
    b.y = xp[kk + 1];
    c = __builtin_amdgcn_wmma_f32_16x16x4_f32(
            /*neg_a=*/false, a, /*neg_b=*/false, b,
            /*c_mod=*/(short)0, c, /*reuse_a=*/false, /*reuse_b=*/false);
  }

  // C/D layout: VGPR r of lanes 0-15 holds M=r, lanes 16-31 hold M=8+r.
  // All 16 columns are identical (B columns equal) -> one lane per half writes.
  float* dst = gvp + (size_t)blockIdx.y * N + o0;
  if (lane == 0) {
    #pragma unroll
    for (int r = 0; r < 8; ++r) dst[r] = c[r];
  } else if (lane == 16) {
    #pragma unroll
    for (int r = 0; r < 8; ++r) dst[8 + r] = c[r];
  }
}

// ---------------------------------------------------------------- reduce
__global__ void __launch_bounds__(256)
reduce_kernel(const float* __restrict__ bias, const float* __restrict__ budp,
              const float* __restrict__ gvp, float* __restrict__ out) {
  int o = blockIdx.x * blockDim.x + threadIdx.x;
  if (o >= N) return;
  float s = bias[o];
  #pragma unroll
  for (int c = 0; c < KC; ++c) s += budp[(size_t)c * N + o];   // fixed order:
  #pragma unroll
  for (int c = 0; c < KS; ++c) s += gvp[(size_t)c * N + o];    // deterministic
  out[o] = s;
}

// ---------------------------------------------------------------- launch
extern "C" void kernel_launch(void* const* d_in, const int* in_sizes, int n_in,
                              void* d_out, int out_size, void* d_ws, size_t ws_size,
                              hipStream_t stream) {
  (void)in_sizes; (void)n_in; (void)out_size; (void)ws_size;
  const float* x    = (const float*)d_in[0];
  const int*   sat  = (const int*)  d_in[1];   // bool -> int32 per harness convention
  const float* wgt  = (const float*)d_in[2];
  const float* bias = (const float*)d_in[3];
  const float* W1   = (const float*)d_in[4];
  const float* b1   = (const float*)d_in[5];
  const float* W2   = (const float*)d_in[6];
  const float* b2   = (const float*)d_in[7];
  const float* W3   = (const float*)d_in[8];
  const float* b3   = (const float*)d_in[9];
  float* out = (float*)d_out;

  // workspace layout (floats): h2m[4N] | xm[N] | budp[KC*N] | gvp[KS*N]  (~1.66MB)
  float*  ws   = (float*)d_ws;
  float4* h2m  = (float4*)ws;
  float*  xm   = ws + (size_t)4 * N;
  float*  budp = ws + (size_t)5 * N;
  float*  gvp  = budp + (size_t)KC * N;

  prep_kernel  <<<N / 256, 256, 0, stream>>>(x, sat, W1, b1, W2, b2, h2m, xm);
  buds_kernel  <<<dim3(N / 256, KC), 256, 0, stream>>>(W3, b3, h2m, budp);
  gemv_kernel  <<<dim3(N / 16, KS), 32, 0, stream>>>(wgt, xm, gvp);
  reduce_kernel<<<N / 256, 256, 0, stream>>>(bias, budp, gvp, out);
}